// DynamicReceptiveSpikeActivation_34969623724321
// MI455X (gfx1250) — compile-verified
//
#include <hip/hip_runtime.h>
#include <cstddef>
#include <cstdint>

// ---------------------------------------------------------------------------
// Streaming LIF spike scan for (B=128, C=256, L=512) fp32.
// 32768 independent rows, sequential scan along L (nonlinear recurrence ->
// no WMMA reformulation). Memory-bound: 128MB traffic / 23.3 TB/s ~ 5.5us.
// CDNA5 path: async global->LDS tile loads (ASYNCcnt) + double buffering,
// coalesced staged stores with non-temporal hints.
// ---------------------------------------------------------------------------

#define BQ        128            // threads per block == rows per block
#define LQ        512            // time steps
#define TC        32             // tile columns (time steps per tile)
#define NT        (LQ / TC)      // 16 tiles
#define INSTRIDE  36             // padded LDS row stride (floats), 144B = 16B aligned
#define OUTSTRIDE 33             // padded LDS row stride for outputs (conflict-free)

#if defined(__HIP_DEVICE_COMPILE__) && __has_builtin(__builtin_amdgcn_global_load_async_to_lds_b128)
#define HAS_ASYNC 1
#else
#define HAS_ASYNC 0
#endif

#if defined(__HIP_DEVICE_COMPILE__) && __has_builtin(__builtin_amdgcn_s_wait_asynccnt)
#define WAIT_ASYNC(n) __builtin_amdgcn_s_wait_asynccnt(n)
#elif defined(__HIP_DEVICE_COMPILE__)
#define WAIT_ASYNC(n) asm volatile("s_wait_asynccnt " #n ::: "memory")
#else
#define WAIT_ASYNC(n)
#endif

#if HAS_ASYNC
// Builtin prototype (from the round-1 diagnostic): param0 is
//   __attribute__((vector_size(16))) int __device__ *   (i.e. v4i in AS1)
typedef int ivec4 __attribute__((vector_size(16)));
typedef __attribute__((address_space(1))) ivec4 g_ivec4;   // global (printed __device__)
typedef __attribute__((address_space(3))) ivec4 l_ivec4;   // LDS   (printed __shared__)
#endif

struct LIF {
  float s_mem, f_mem, n_mem, px;
  float s_prev, f_prev, n_prev, n_h2;
};

// One time step for one row. Returns the OUTPUT for the PREVIOUS step
// (out[t-1] needs n_spk[t] for the 3-tap burst stencil).
__device__ __forceinline__ float lif_step(LIF& st, float x) {
  float s_mem = fmaf(0.85f, st.s_mem, x) - st.s_prev * 0.5f;
  float s_spk = (s_mem > 0.5f) ? 1.0f : 0.0f;
  float delta = fabsf(x - st.px);
  float f_mem = fmaf(0.70f, st.f_mem, delta) - st.f_prev * 0.9f;
  float f_spk = (f_mem > 0.9f) ? 1.0f : 0.0f;
  float n_mem = fmaf(0.95f, st.n_mem, x) - st.n_prev * 3.0f;
  float n_spk = (n_mem > 3.0f) ? 1.0f : 0.0f;
  // burst[t-1] = (n[t-2] + n[t-1] + n[t]) > 0   (spikes are in {0,1})
  float burst = ((st.n_h2 + st.n_prev + n_spk) > 0.0f) ? 1.0f : 0.0f;
  float outv = st.s_prev + st.f_prev + burst;
  st.s_mem = s_mem; st.f_mem = f_mem; st.n_mem = n_mem; st.px = x;
  st.n_h2 = st.n_prev;
  st.s_prev = s_spk; st.f_prev = f_spk; st.n_prev = n_spk;
  return outv;
}

#if HAS_ASYNC
// Issue 8 per-lane b128 async copies: tile of 128 rows x 32 cols, coalesced
// (a wave covers 4 rows x 8 chunks = 4 contiguous 128B segments).
__device__ __forceinline__ void prefetch_tile_async(const float* gbase, float* ldsbuf,
                                                    int tid, int k) {
#pragma unroll
  for (int j = 0; j < 8; ++j) {
    int n = tid + BQ * j;          // 0..1023
    int row = n >> 3;              // 0..127
    int ch  = n & 7;               // 16B chunk within the 128B tile row
    const float* gp = gbase + (size_t)row * LQ + k * TC + ch * 4;
    float* lp = ldsbuf + row * INSTRIDE + ch * 4;   // 16B-aligned LDS dest
    __builtin_amdgcn_global_load_async_to_lds_b128(
        (g_ivec4*)gp, (l_ivec4*)lp, /*offset=*/0, /*cpol=*/0);
  }
}
#else
__device__ __forceinline__ void copy_tile_sync(const float* gbase, float* ldsbuf,
                                               int tid, int k) {
#pragma unroll
  for (int j = 0; j < 8; ++j) {
    int n = tid + BQ * j;
    int row = n >> 3;
    int ch  = n & 7;
    float4 v = *(const float4*)(gbase + (size_t)row * LQ + k * TC + ch * 4);
    *(float4*)(ldsbuf + row * INSTRIDE + ch * 4) = v;
  }
}
#endif

__global__ __launch_bounds__(BQ) void snn_scan_kernel(const float* __restrict__ in,
                                                      float* __restrict__ out) {
#if HAS_ASYNC
  __shared__ float lds_in[2 * BQ * INSTRIDE];   // double-buffered input tiles
#else
  __shared__ float lds_in[BQ * INSTRIDE];
#endif
  __shared__ float lds_out[BQ * OUTSTRIDE];     // slot s of tile k <-> column k*32-1+s

  const int tid = threadIdx.x;
  const size_t row0 = (size_t)blockIdx.x * BQ;
  const float* gin = in + row0 * LQ;
  float* gout = out + row0 * LQ;

  LIF st;
  st.s_mem = st.f_mem = st.n_mem = st.px = 0.0f;
  st.s_prev = st.f_prev = st.n_prev = st.n_h2 = 0.0f;

#if HAS_ASYNC
  prefetch_tile_async(gin, lds_in, tid, 0);
#endif

  for (int k = 0; k < NT; ++k) {
    float* inb;
#if HAS_ASYNC
    inb = lds_in + (k & 1) * (BQ * INSTRIDE);
    if (k + 1 < NT) {
      prefetch_tile_async(gin, lds_in + ((k + 1) & 1) * (BQ * INSTRIDE), tid, k + 1);
      WAIT_ASYNC(8);      // tile k (issued last iter) fully landed; k+1 in flight
    } else {
      WAIT_ASYNC(0);      // drain final tile
    }
#else
    inb = lds_in;
    copy_tile_sync(gin, lds_in, tid, k);
#endif
    __syncthreads();      // all waves' portions of tile k visible in LDS

    // --- sequential scan over this tile; one row per thread ---------------
    const float4* rowp = (const float4*)(inb + tid * INSTRIDE);
    float* orow = lds_out + tid * OUTSTRIDE;
#pragma unroll
    for (int q = 0; q < 8; ++q) {
      float4 xv = rowp[q];
      orow[q * 4 + 0] = lif_step(st, xv.x);
      orow[q * 4 + 1] = lif_step(st, xv.y);
      orow[q * 4 + 2] = lif_step(st, xv.z);
      orow[q * 4 + 3] = lif_step(st, xv.w);
    }
    __syncthreads();      // out tile complete; input tile k fully consumed

    // --- coalesced staged store (columns shifted by -1) -------------------
#pragma unroll 4
    for (int j = 0; j < 32; ++j) {
      int n = tid + BQ * j;        // 0..4095
      int row = n >> 5;            // 0..127
      int s = n & 31;
      int col = k * TC - 1 + s;    // global time index
      if (col >= 0)
        __builtin_nontemporal_store(lds_out[row * OUTSTRIDE + s],
                                    gout + (size_t)row * LQ + col);
    }
    // next iteration's first __syncthreads() separates these lds_out reads
    // from the next compute phase's writes
  }

  // Drain: out[L-1] uses n[L] == 0.
  float tail = st.s_prev + st.f_prev +
               (((st.n_h2 + st.n_prev) > 0.0f) ? 1.0f : 0.0f);
  gout[(size_t)tid * LQ + (LQ - 1)] = tail;
}

// ---------------------------------------------------------------------------
extern "C" void kernel_launch(void* const* d_in, const int* in_sizes, int n_in,
                              void* d_out, int out_size, void* d_ws, size_t ws_size,
                              hipStream_t stream) {
  (void)n_in; (void)out_size; (void)d_ws; (void)ws_size;
  const float* act = (const float*)d_in[0];
  float* out = (float*)d_out;
  const int rows = in_sizes[0] / LQ;      // 32768
  const int blocks = rows / BQ;           // 256
  snn_scan_kernel<<<blocks, BQ, 0, stream>>>(act, out);
}